// SAModule1_84817014162236
// MI455X (gfx1250) — compile-verified
//
#include <hip/hip_runtime.h>
#include <hip/hip_bf16.h>

#define NPTS 16384
#define MQ   4096
#define KNB  64
#define DF   64
#define FPAD 80          // 67 feature cols padded to 80 (multiple of 4)
#define H1C  64
#define H2C  128
#define CAP  2048
#define BIGF 1e30f
#define R2S  0.01f       // 0.1^2
#define R2B  0.04f       // 0.2^2
#define FPS_T 1024
#define PPT  (NPTS / FPS_T)   // 16 points per thread

typedef float v2f __attribute__((ext_vector_type(2)));
typedef float v8f __attribute__((ext_vector_type(8)));

// ---------------------------------------------------------------------------
// 1) Farthest point sampling: one persistent workgroup, state in registers.
//    Emits qpos (= pos[idx]) directly into the d_out qpos region.
// ---------------------------------------------------------------------------
__global__ __launch_bounds__(FPS_T) void fps_kernel(
    const float* __restrict__ pos, float* __restrict__ qpos_out) {
  const int t    = threadIdx.x;
  const int lane = t & 31;
  const int wv   = t >> 5;

  float px[PPT], py[PPT], pz[PPT], md[PPT];
#pragma unroll
  for (int i = 0; i < PPT; ++i) {
    int p = t + i * FPS_T;
    px[i] = pos[3 * p + 0];
    py[i] = pos[3 * p + 1];
    pz[i] = pos[3 * p + 2];
    md[i] = BIGF;
  }

  __shared__ float s_cur[3];
  __shared__ float s_val[32];
  __shared__ int   s_idx[32];
  __shared__ int   s_next;

  int cur = 0;
  for (int m = 0; m < MQ; ++m) {
    if ((cur & (FPS_T - 1)) == t) {
      int i = cur >> 10;
      s_cur[0] = px[i]; s_cur[1] = py[i]; s_cur[2] = pz[i];
      qpos_out[3 * m + 0] = px[i];
      qpos_out[3 * m + 1] = py[i];
      qpos_out[3 * m + 2] = pz[i];
    }
    __syncthreads();
    float cx = s_cur[0], cy = s_cur[1], cz = s_cur[2];

    float bv = -1.0f; int bi = 0x7fffffff;
#pragma unroll
    for (int i = 0; i < PPT; ++i) {
      float dx = px[i] - cx, dy = py[i] - cy, dz = pz[i] - cz;
      float d  = dx * dx + dy * dy + dz * dz;
      float nm = fminf(md[i], d);
      md[i] = nm;
      int p = t + i * FPS_T;
      if (nm > bv || (nm == bv && p < bi)) { bv = nm; bi = p; }
    }
    // wave32 lexicographic (max val, min idx) reduce
    for (int off = 16; off > 0; off >>= 1) {
      float ov = __shfl_down(bv, off);
      int   oi = __shfl_down(bi, off);
      if (ov > bv || (ov == bv && oi < bi)) { bv = ov; bi = oi; }
    }
    if (lane == 0) { s_val[wv] = bv; s_idx[wv] = bi; }
    __syncthreads();
    if (wv == 0) {
      float v = s_val[lane]; int ii = s_idx[lane];
      for (int off = 16; off > 0; off >>= 1) {
        float ov = __shfl_down(v, off);
        int   oi = __shfl_down(ii, off);
        if (ov > v || (ov == v && oi < ii)) { v = ov; ii = oi; }
      }
      if (lane == 0) s_next = ii;
    }
    __syncthreads();
    cur = s_next;
  }
}

// ---------------------------------------------------------------------------
// 2) Ball query: one block per centroid. One scan collects r=0.2 candidates
//    into LDS, then 64-round min-extraction per radius (K nearest in-ball).
// ---------------------------------------------------------------------------
__global__ __launch_bounds__(256) void ballquery_kernel(
    const float* __restrict__ pos, const float* __restrict__ qpos,
    int* __restrict__ nbr) {
  const int q    = blockIdx.x;
  const int tid  = threadIdx.x;
  const int lane = tid & 31;
  const int wv   = tid >> 5;

  __shared__ float         cd2[CAP];
  __shared__ int           cid[CAP];
  __shared__ unsigned char taken[CAP];
  __shared__ int   s_cnt;
  __shared__ float rv[8];
  __shared__ int   rp[8];
  __shared__ int   rs[8];

  if (tid == 0) s_cnt = 0;
  __syncthreads();

  const float qx = qpos[3 * q + 0], qy = qpos[3 * q + 1], qz = qpos[3 * q + 2];
  for (int p = tid; p < NPTS; p += 256) {
    float dx = pos[3 * p + 0] - qx;
    float dy = pos[3 * p + 1] - qy;
    float dz = pos[3 * p + 2] - qz;
    float d2 = dx * dx + dy * dy + dz * dz;
    if (d2 <= R2B) {
      int slot = atomicAdd(&s_cnt, 1);
      if (slot < CAP) { cd2[slot] = d2; cid[slot] = p; }
    }
  }
  __syncthreads();
  int n = s_cnt; if (n > CAP) n = CAP;

  for (int rad = 0; rad < 2; ++rad) {
    const float r2 = (rad == 0) ? R2S : R2B;
    for (int i = tid; i < n; i += 256) taken[i] = 0;
    __syncthreads();
    int* nrow = nbr + ((size_t)rad * MQ + q) * KNB;
    for (int k = 0; k < KNB; ++k) {
      float bv = BIGF; int bp = 0x7fffffff; int bs = -1;
      for (int i = tid; i < n; i += 256) {
        if (!taken[i] && cd2[i] <= r2) {
          float v = cd2[i]; int pp = cid[i];
          if (v < bv || (v == bv && pp < bp)) { bv = v; bp = pp; bs = i; }
        }
      }
      for (int off = 16; off > 0; off >>= 1) {
        float ov = __shfl_down(bv, off);
        int   op = __shfl_down(bp, off);
        int   os = __shfl_down(bs, off);
        if (ov < bv || (ov == bv && op < bp)) { bv = ov; bp = op; bs = os; }
      }
      if (lane == 0) { rv[wv] = bv; rp[wv] = bp; rs[wv] = bs; }
      __syncthreads();
      if (tid == 0) {
        float fv = rv[0]; int fp = rp[0]; int fs = rs[0];
        for (int w = 1; w < 8; ++w)
          if (rv[w] < fv || (rv[w] == fv && rp[w] < fp)) { fv = rv[w]; fp = rp[w]; fs = rs[w]; }
        if (fs >= 0) { taken[fs] = 1; nrow[k] = fp; }
        else         { nrow[k] = -1; }
      }
      __syncthreads();
    }
  }
}

// ---------------------------------------------------------------------------
// 3) Fused gather + MLP(67->64->128, ReLU) + masked max, per (radius, centroid).
//    f32 WMMA 16x16x4 tiles; A from LDS, B streamed from L2-resident weights.
// ---------------------------------------------------------------------------
__global__ __launch_bounds__(128) void pointconv_kernel(
    const float* __restrict__ x, const float* __restrict__ pos,
    const float* __restrict__ qpos, const int* __restrict__ nbr,
    const float* __restrict__ W1, const float* __restrict__ b1,
    const float* __restrict__ W2, const float* __restrict__ b2,
    float* __restrict__ out) {
  const int q    = blockIdx.x & (MQ - 1);
  const int rad  = blockIdx.x >> 12;
  const int tid  = threadIdx.x;
  const int lane = tid & 31;
  const int wv   = tid >> 5;

  __shared__ float sBuf[KNB * H2C];   // feat (ld FPAD) aliased with h2 (ld H2C)
  __shared__ float sH1[KNB * 72];     // h1, padded ld 72
  __shared__ int   sNb[KNB];

  const int* nb = nbr + ((size_t)rad * MQ + q) * KNB;
  if (tid < KNB) sNb[tid] = nb[tid];
  __syncthreads();

  const float qx = qpos[3 * q + 0], qy = qpos[3 * q + 1], qz = qpos[3 * q + 2];

  // build feat[64 x 67] -> padded FPAD, invalid rows zeroed (safe under ReLU+max)
  {
    int j  = tid >> 1;
    int c0 = (tid & 1) * (FPAD / 2);
    int p  = sNb[j];
    float* frow = sBuf + j * FPAD;
    for (int c = c0; c < c0 + FPAD / 2; ++c) {
      float v = 0.0f;
      if (p >= 0) {
        if (c < DF)       v = x[(size_t)p * DF + c];
        else if (c == 64) v = pos[3 * p + 0] - qx;
        else if (c == 65) v = pos[3 * p + 1] - qy;
        else if (c == 66) v = pos[3 * p + 2] - qz;
      }
      frow[c] = v;
    }
  }
  __syncthreads();

  const int ksel = (lane >> 4) << 1;     // 0 or 2 (K split across lane halves)
  const int ncol = lane & 15;
  const v8f vzero = {0.f, 0.f, 0.f, 0.f, 0.f, 0.f, 0.f, 0.f};

  // ---- layer 1: h1 = relu(feat @ W1 + b1), 64x64; wave wv owns col tile wv ----
  {
    const int col = 16 * wv + ncol;
    v8f acc[4] = {vzero, vzero, vzero, vzero};
    for (int kk = 0; kk < FPAD; kk += 4) {
      int kr = kk + ksel;
      v2f bf;
      bf.x = (kr     < 67) ? W1[kr * H1C + col]       : 0.0f;
      bf.y = (kr + 1 < 67) ? W1[(kr + 1) * H1C + col] : 0.0f;
#pragma unroll
      for (int rt = 0; rt < 4; ++rt) {
        int row = 16 * rt + ncol;
        v2f af;
        af.x = sBuf[row * FPAD + kr];
        af.y = sBuf[row * FPAD + kr + 1];
        acc[rt] = __builtin_amdgcn_wmma_f32_16x16x4_f32(
            false, af, false, bf, (short)0, acc[rt], false, false);
      }
    }
    float bb = b1[col];
#pragma unroll
    for (int rt = 0; rt < 4; ++rt)
#pragma unroll
      for (int g = 0; g < 8; ++g) {
        int row = 16 * rt + g + 8 * (lane >> 4);
        sH1[row * 72 + col] = fmaxf(acc[rt][g] + bb, 0.0f);
      }
  }
  __syncthreads();

  // ---- layer 2: h2 = relu(h1 @ W2 + b2), 64x128; wave wv owns col tiles wv, wv+4 ----
  {
    v8f acc[2][4] = {{vzero, vzero, vzero, vzero}, {vzero, vzero, vzero, vzero}};
    for (int kk = 0; kk < H1C; kk += 4) {
      int kr = kk + ksel;
      v2f bf[2];
#pragma unroll
      for (int cs = 0; cs < 2; ++cs) {
        int col = 16 * (wv + 4 * cs) + ncol;
        bf[cs].x = W2[kr * H2C + col];
        bf[cs].y = W2[(kr + 1) * H2C + col];
      }
#pragma unroll
      for (int rt = 0; rt < 4; ++rt) {
        int row = 16 * rt + ncol;
        v2f af;
        af.x = sH1[row * 72 + kr];
        af.y = sH1[row * 72 + kr + 1];
#pragma unroll
        for (int cs = 0; cs < 2; ++cs)
          acc[cs][rt] = __builtin_amdgcn_wmma_f32_16x16x4_f32(
              false, af, false, bf[cs], (short)0, acc[cs][rt], false, false);
      }
    }
#pragma unroll
    for (int cs = 0; cs < 2; ++cs) {
      int col = 16 * (wv + 4 * cs) + ncol;
      float bb = b2[col];
#pragma unroll
      for (int rt = 0; rt < 4; ++rt)
#pragma unroll
        for (int g = 0; g < 8; ++g) {
          int row = 16 * rt + g + 8 * (lane >> 4);
          float v = fmaxf(acc[cs][rt][g] + bb, 0.0f);
          sBuf[row * H2C + col] = (sNb[row] >= 0) ? v : 0.0f;
        }
    }
  }
  __syncthreads();

  // ---- masked max over 64 neighbor rows (all values >= 0; self is always valid) ----
  {
    float m = 0.0f;
    for (int j = 0; j < KNB; ++j) m = fmaxf(m, sBuf[j * H2C + tid]);
    out[(size_t)q * 256 + rad * H2C + tid] = m;
  }
}

// ---------------------------------------------------------------------------
// 4) Zero tail of d_out (batch[idx] output: int64 zeros == zero bit pattern)
// ---------------------------------------------------------------------------
__global__ void fill_zero_kernel(float* __restrict__ p, int nelem) {
  int i = blockIdx.x * blockDim.x + threadIdx.x;
  if (i < nelem) p[i] = 0.0f;
}

extern "C" void kernel_launch(void* const* d_in, const int* in_sizes, int n_in,
                              void* d_out, int out_size, void* d_ws, size_t ws_size,
                              hipStream_t stream) {
  const float* x   = (const float*)d_in[0];
  const float* pos = (const float*)d_in[1];
  // d_in[2] = batch (all-zero int64), not needed
  const float* W1  = (const float*)d_in[3];
  const float* b1  = (const float*)d_in[4];
  const float* W2  = (const float*)d_in[5];
  const float* b2  = (const float*)d_in[6];

  float* out  = (float*)d_out;
  float* qpos = out + (size_t)MQ * 256;     // qpos region of d_out, written by fps
  int*   nbr  = (int*)d_ws;                 // [2][MQ][KNB] neighbor indices

  fps_kernel<<<1, FPS_T, 0, stream>>>(pos, qpos);
  ballquery_kernel<<<MQ, 256, 0, stream>>>(pos, qpos, nbr);
  pointconv_kernel<<<2 * MQ, 128, 0, stream>>>(x, pos, qpos, nbr, W1, b1, W2, b2, out);

  int tail_begin = MQ * 256 + MQ * 3;
  int tail = out_size - tail_begin;
  if (tail > 0)
    fill_zero_kernel<<<(tail + 255) / 256, 256, 0, stream>>>(out + tail_begin, tail);
}